// Hadmar_29961691857287
// MI455X (gfx1250) — compile-verified
//
#include <hip/hip_runtime.h>
#include <hip/hip_bf16.h>

// y = (1/512) * H512 @ X @ H512,  X: [128, 512, 512] f32.
// H[i,j] = (-1)^popcount(i&j)  -> synthesized in registers, never loaded.
//
// Pass 1: per (b, 16-row block): Y1 = Xblock[16x512] @ H[512x512]  (row/D transform)
// Pass 2: per (b, 16-col slab) : Y  = H[512x512] @ Y1slab[512x16], scaled, in place.
// Memory-bound by ~100x: 256 MB HBM floor (~11 us at 23.3 TB/s) vs ~69 GFLOP f32 WMMA.
// The 128 MB intermediate is kept L2-resident (192 MB L2): input reads are NT,
// final output stores are NT, intermediate traffic is RT.
// LDS fills use CDNA5 async global->LDS DMA (ASYNCcnt), bypassing VGPRs.

typedef float v2f __attribute__((ext_vector_type(2)));
typedef float v4f __attribute__((ext_vector_type(4)));
typedef float v8f __attribute__((ext_vector_type(8)));

__device__ __forceinline__ float had_sign(int a, int b) {
    // (-1)^popcount(a & b)
    return (__builtin_popcount(a & b) & 1) ? -1.0f : 1.0f;
}

// ---- CDNA5 async global->LDS copy (16B per lane), inline asm per CDNA5 ISA ----
// Generic pointers into __shared__ carry the LDS offset in their low 32 bits on AMDGPU.
__device__ __forceinline__ void async_copy16_nt(unsigned lds_addr, const void* gaddr) {
    asm volatile("global_load_async_to_lds_b128 %0, %1, off th:TH_LOAD_NT"
                 :: "v"(lds_addr), "v"(gaddr) : "memory");
}
__device__ __forceinline__ void async_copy16(unsigned lds_addr, const void* gaddr) {
    asm volatile("global_load_async_to_lds_b128 %0, %1, off"
                 :: "v"(lds_addr), "v"(gaddr) : "memory");
}
__device__ __forceinline__ void wait_async_lds() {
    asm volatile("s_wait_asynccnt 0x0" ::: "memory");
}

// ---------------- Pass 1: row transform  Y1 = X @ H ----------------
__global__ __launch_bounds__(128) void had_pass1(const float* __restrict__ x,
                                                 float* __restrict__ y) {
    __shared__ float tile[16 * 512];                 // 32 KB (of 320 KB/WGP)
    const int b  = blockIdx.x >> 5;                  // batch
    const int n0 = (blockIdx.x & 31) << 4;           // 16-row block

    // Cooperative async fill: the 16x512 block is fully contiguous (32 KB).
    // NT: x is read exactly once; keep L2 for the intermediate.
    const float* src = x + ((size_t)b * 512 + n0) * 512;
    const unsigned lbase = (unsigned)(size_t)&tile[0];
    for (int u = threadIdx.x; u < 2048; u += 128)
        async_copy16_nt(lbase + (unsigned)u * 16u, src + (size_t)u * 4);
    wait_async_lds();
    __syncthreads();

    const int lane  = threadIdx.x & 31;
    const int wave  = threadIdx.x >> 5;
    const int m     = lane & 15;                     // row (A) / col (B,C) within tile
    const int khalf = (lane >> 4) << 1;              // lanes 16-31 hold K+2,K+3

    v8f c[8];
    #pragma unroll
    for (int t = 0; t < 8; ++t)
        #pragma unroll
        for (int j = 0; j < 8; ++j) c[t][j] = 0.0f;

    // A = X fragment (16x4, from LDS), B = H fragment (4x16, synthesized).
    for (int k0 = 0; k0 < 512; k0 += 4) {
        const int kk = k0 + khalf;
        const v2f a = *(const v2f*)&tile[m * 512 + kk];   // ds_load_b64
        #pragma unroll
        for (int t = 0; t < 8; ++t) {
            const int n = (((wave << 3) + t) << 4) + m;   // output column
            v2f bb;
            bb.x = had_sign(kk,     n);
            bb.y = had_sign(kk + 1, n);
            c[t] = __builtin_amdgcn_wmma_f32_16x16x4_f32(
                       false, a, false, bb, (short)0, c[t], false, false);
        }
    }

    // Store C tiles: VGPR r, lanes 0-15 -> M=r, lanes 16-31 -> M=r+8; N = lane&15.
    // RT stores -> intermediate stays resident in the 192 MB L2 for pass 2.
    const int rbase = (lane >> 4) << 3;
    float* dstb = y + ((size_t)b * 512 + n0) * 512;
    #pragma unroll
    for (int t = 0; t < 8; ++t) {
        float* dst = dstb + (((wave << 3) + t) << 4) + m;
        #pragma unroll
        for (int r = 0; r < 8; ++r)
            dst[(size_t)(rbase + r) * 512] = c[t][r];
    }
}

// ---------------- Pass 2: column transform  Y = H @ Y1, scaled, in place ----------------
__global__ __launch_bounds__(128) void had_pass2(float* __restrict__ y) {
    __shared__ float tile[512 * 16];                 // 32 KB, row stride 16 floats
    const int b  = blockIdx.x >> 5;
    const int d0 = (blockIdx.x & 31) << 4;           // 16-column slab

    float* base = y + (size_t)b * 512 * 512 + d0;
    // Async fill: 512 rows x 16 cols (one b128 per row-quarter), mostly L2 hits (RT).
    const unsigned lbase = (unsigned)(size_t)&tile[0];
    for (int u = threadIdx.x; u < 2048; u += 128) {
        const int row = u >> 2, q = (u & 3) << 2;
        async_copy16(lbase + (unsigned)(row * 16 + q) * 4u,
                     base + (size_t)row * 512 + q);
    }
    wait_async_lds();
    __syncthreads();                                  // all reads done -> in-place safe

    const int lane  = threadIdx.x & 31;
    const int wave  = threadIdx.x >> 5;
    const int m     = lane & 15;
    const int khalf = (lane >> 4) << 1;

    v8f c[8];
    #pragma unroll
    for (int t = 0; t < 8; ++t)
        #pragma unroll
        for (int j = 0; j < 8; ++j) c[t][j] = 0.0f;

    // A = H fragment (16x4, synthesized), B = Y1 fragment (4x16, from LDS).
    for (int k0 = 0; k0 < 512; k0 += 4) {
        const int kk = k0 + khalf;
        v2f bb;
        bb.x = tile[kk * 16 + m];                     // rows map to disjoint 16-bank
        bb.y = tile[(kk + 1) * 16 + m];               // groups: conflict-free
        #pragma unroll
        for (int t = 0; t < 8; ++t) {
            const int nrow = (((wave << 3) + t) << 4) + m;  // output row
            v2f a;
            a.x = had_sign(nrow, kk);
            a.y = had_sign(nrow, kk + 1);
            c[t] = __builtin_amdgcn_wmma_f32_16x16x4_f32(
                       false, a, false, bb, (short)0, c[t], false, false);
        }
    }

    const float scale = 1.0f / 512.0f;                // 1/sqrt(512*512)
    const int rbase = (lane >> 4) << 3;
    float* dstb = y + (size_t)b * 512 * 512 + d0;
    #pragma unroll
    for (int t = 0; t < 8; ++t) {
        #pragma unroll
        for (int r = 0; r < 8; ++r) {
            const int nrow = (((wave << 3) + t) << 4) + rbase + r;
            // Final output: non-temporal, never re-read; adjacent WGs' 64B strips
            // coalesce into full lines in L2 before HBM writeback.
            __builtin_nontemporal_store(c[t][r] * scale,
                                        dstb + (size_t)nrow * 512 + m);
        }
    }
}

extern "C" void kernel_launch(void* const* d_in, const int* in_sizes, int n_in,
                              void* d_out, int out_size, void* d_ws, size_t ws_size,
                              hipStream_t stream) {
    const float* x = (const float*)d_in[0];
    float* out = (float*)d_out;
    (void)in_sizes; (void)n_in; (void)out_size; (void)d_ws; (void)ws_size;
    // 128 batches x 32 blocks of 16 rows/cols = 4096 workgroups, 4 waves (wave32) each.
    had_pass1<<<dim3(4096), dim3(128), 0, stream>>>(x, out);
    had_pass2<<<dim3(4096), dim3(128), 0, stream>>>(out);
}